// DetectorKe_1924145349176
// MI455X (gfx1250) — compile-verified
//
#include <hip/hip_runtime.h>
#include <hip/hip_bf16.h>

typedef __attribute__((ext_vector_type(2))) float v2f;
typedef __attribute__((ext_vector_type(8))) float v8f;

// ---------------------------------------------------------------------------
// Setup kernel: one block of 1024 threads = 32 waves, wave m handles center m.
//   * |det(L_m)| via unrolled Gaussian elimination, rows in registers,
//     pivot rows broadcast with __shfl (wave32 lockstep).
//   * nb_m[k] = -sum_d L_m[d][k] * c_m[d]    -> ws[32 + m*32 + k]  (pre-negated)
//   * coefs[m] = softmax(w)[m] * |det(L_m)|  -> ws[m]
// ---------------------------------------------------------------------------
__global__ __launch_bounds__(1024) void gmm_setup_kernel(
    const float* __restrict__ L,        // (32,32,32)
    const float* __restrict__ centers,  // (32,32)
    const float* __restrict__ weights,  // (32)
    float* __restrict__ ws)             // [0,32) coefs, [32,32+1024) -b
{
    __shared__ float detSh[32];
    const int tid  = threadIdx.x;
    const int m    = tid >> 5;
    const int lane = tid & 31;

    const float* Lm = L + m * 1024;

    // nb_m[lane] = -sum_d L[d][lane] * c[d]
    {
        const float* cm = centers + m * 32;
        float bk = 0.0f;
        #pragma unroll
        for (int d = 0; d < 32; ++d)
            bk = fmaf(Lm[d * 32 + lane], cm[d], bk);
        ws[32 + m * 32 + lane] = -bk;
    }

    // Gaussian elimination: lane owns row `lane` of L_m, kept in registers.
    float row[32];
    #pragma unroll
    for (int j = 0; j < 32; ++j) row[j] = Lm[lane * 32 + j];

    float det = 1.0f;
    #pragma unroll
    for (int i = 0; i < 32; ++i) {
        const float pv = __shfl(row[i], i, 32);   // pivot value (row i is final)
        det *= pv;
        const float f = (lane > i) ? (row[i] / pv) : 0.0f;
        #pragma unroll
        for (int j = i; j < 32; ++j) {
            const float prj = __shfl(row[j], i, 32);
            row[j] = fmaf(-f, prj, row[j]);
        }
    }
    if (lane == 0) detSh[m] = fabsf(det);
    __syncthreads();

    // Wave 0: softmax over 32 weights, scale by |det|.
    if (m == 0) {
        const float w = weights[lane];
        float mx = w;
        #pragma unroll
        for (int off = 16; off > 0; off >>= 1)
            mx = fmaxf(mx, __shfl_xor(mx, off, 32));
        const float e = expf(w - mx);
        float s = e;
        #pragma unroll
        for (int off = 16; off > 0; off >>= 1)
            s += __shfl_xor(s, off, 32);
        ws[lane] = (e / s) * detSh[lane];
    }
}

// ---------------------------------------------------------------------------
// Main kernel: 256 threads = 8 waves; wave handles 32 points (2 row tiles).
// Z^T = L^T @ X^T via V_WMMA_F32_16X16X4_F32.
// L is staged in LDS in *fragment order* so every A-operand is one aligned
// ds_load_b64 (conflict-free: 32 lanes x 8B contiguous = 64 banks once).
// Accumulator is initialized to -b (pre-negated in ws), so the GEMM directly
// yields z-b; dist = in-lane sum of squares + one shfl_xor(16).
// ---------------------------------------------------------------------------
__global__ __launch_bounds__(256) void gmm_main_kernel(
    const float* __restrict__ X,      // (N,32)
    const float* __restrict__ L,      // (32,32,32)
    const float* __restrict__ ws,     // coefs[32], -b[32*32]
    float* __restrict__ out)
{
    // Fragment-ordered L: Lf[((m*8 + j)*2 + kt)*32 + lane] = v2f pair
    __shared__ float Lsh[32 * 32 * 32];   // 128 KB
    __shared__ float nbsh[32 * 32];       // 4 KB, holds -b
    __shared__ float csh[32];

    const int tid = threadIdx.x;
    // Cooperative fill with on-the-fly reorder into fragment layout.
    for (int i = tid; i < 32 * 32 * 32; i += 256) {
        const int t     = i & 1;          // element within the v2f pair
        const int p     = i >> 1;         // pair index
        const int flane = p & 31;
        const int kt    = (p >> 5) & 1;
        const int j     = (p >> 6) & 7;
        const int m     = p >> 9;
        const int fhi   = flane >> 4;
        const int fcol  = flane & 15;
        const int d     = 4 * j + 2 * fhi + t;
        const int k     = 16 * kt + fcol;
        Lsh[i] = L[m * 1024 + d * 32 + k];
    }
    for (int i = tid; i < 32 * 32; i += 256) nbsh[i] = ws[32 + i];
    if (tid < 32) csh[tid] = ws[tid];
    __syncthreads();

    const int wave = tid >> 5;
    const int lane = tid & 31;
    const int col  = lane & 15;     // point column / k column within tile
    const int hi   = lane >> 4;     // half-wave select

    const long base = (long)blockIdx.x * 256 + wave * 32;

    // B-operand fragments: X^T, tile0 = points base..base+15, tile1 = +16..+31.
    v2f xb0[8], xb1[8];
    const float* xr0 = X + (base + col) * 32;
    const float* xr1 = X + (base + 16 + col) * 32;
    #pragma unroll
    for (int j = 0; j < 8; ++j) {
        const int d = 4 * j + 2 * hi;
        xb0[j] = *(const v2f*)(xr0 + d);
        xb1[j] = *(const v2f*)(xr1 + d);
    }

    float res0 = 0.0f, res1 = 0.0f;

    for (int m = 0; m < 32; ++m) {
        // Fragment base for this m (in v2f units), lane-indexed.
        const v2f* Lf = (const v2f*)Lsh + (m * 16) * 32 + lane;
        const float* nbm = nbsh + m * 32;

        // Accumulators: D[k_local, point] with k_local = v + 8*hi; init -b.
        v8f acc00, acc01, acc10, acc11;   // [ktile][point-tile]
        #pragma unroll
        for (int v = 0; v < 8; ++v) {
            const float b0 = nbm[v + 8 * hi];        // -b[k],   k = v+8hi
            const float b1 = nbm[16 + v + 8 * hi];   // -b[k+16]
            acc00[v] = b0; acc01[v] = b0;
            acc10[v] = b1; acc11[v] = b1;
        }

        #pragma unroll
        for (int j = 0; j < 8; ++j) {
            const v2f la = Lf[(j * 2 + 0) * 32];   // ktile 0, single ds_load_b64
            const v2f ha = Lf[(j * 2 + 1) * 32];   // ktile 1

            acc00 = __builtin_amdgcn_wmma_f32_16x16x4_f32(false, la, false, xb0[j], (short)0, acc00, false, false);
            acc01 = __builtin_amdgcn_wmma_f32_16x16x4_f32(false, la, false, xb1[j], (short)0, acc01, false, false);
            acc10 = __builtin_amdgcn_wmma_f32_16x16x4_f32(false, ha, false, xb0[j], (short)0, acc10, false, false);
            acc11 = __builtin_amdgcn_wmma_f32_16x16x4_f32(false, ha, false, xb1[j], (short)0, acc11, false, false);
        }

        // dist = sum_k (z_k - b_k)^2 : 16 k's in-lane, other 16 in lane^16.
        float s0 = 0.0f, s1 = 0.0f;
        #pragma unroll
        for (int v = 0; v < 8; ++v) {
            s0 = fmaf(acc00[v], acc00[v], s0);
            s0 = fmaf(acc10[v], acc10[v], s0);
            s1 = fmaf(acc01[v], acc01[v], s1);
            s1 = fmaf(acc11[v], acc11[v], s1);
        }
        s0 += __shfl_xor(s0, 16, 32);
        s1 += __shfl_xor(s1, 16, 32);

        const float cm = csh[m];
        res0 = fmaf(cm, __expf(-s0), res0);
        res1 = fmaf(cm, __expf(-s1), res1);
    }

    // Result for point base+col in res0, base+16+col in res1 (both halves hold it).
    if (hi == 0) {
        out[base + col]      = res0;
        out[base + 16 + col] = res1;
    }
}

extern "C" void kernel_launch(void* const* d_in, const int* in_sizes, int n_in,
                              void* d_out, int out_size, void* d_ws, size_t ws_size,
                              hipStream_t stream) {
    const float* points  = (const float*)d_in[0];   // (N,32) f32
    const float* centers = (const float*)d_in[1];   // (32,32) f32
    const float* L       = (const float*)d_in[2];   // (32,32,32) f32
    const float* weights = (const float*)d_in[3];   // (32) f32
    float* out = (float*)d_out;
    float* ws  = (float*)d_ws;                      // needs 4224 bytes

    const int N = in_sizes[0] / 32;                 // 131072

    gmm_setup_kernel<<<1, 1024, 0, stream>>>(L, centers, weights, ws);
    gmm_main_kernel<<<N / 256, 256, 0, stream>>>(points, L, ws, out);
}